// HeteroGraphEncoder_74895639708001
// MI455X (gfx1250) — compile-verified
//
#include <hip/hip_runtime.h>

constexpr int HDIM = 128;

typedef __attribute__((ext_vector_type(2))) float v2f;
typedef __attribute__((ext_vector_type(8))) float v8f;
typedef int v4i __attribute__((vector_size(16)));

// ---------------------------------------------------------------------------
// CDNA5 async global->LDS staging (ASYNCcnt path) with portable fallback.
// Builtin prototype (probed from diagnostic): (v4i as1*, v4i as3*, imm, imm)
// ---------------------------------------------------------------------------
#if defined(__has_builtin)
#if __has_builtin(__builtin_amdgcn_global_load_async_to_lds_b128)
#define USE_ASYNC_LDS 1
#endif
#endif

#define GLOBAL_AS __attribute__((address_space(1)))
#define LDS_AS __attribute__((address_space(3)))

__device__ __forceinline__ void g2lds_f4(const float* g, float* l) {
#ifdef USE_ASYNC_LDS
  __builtin_amdgcn_global_load_async_to_lds_b128((GLOBAL_AS v4i*)g,
                                                 (LDS_AS v4i*)l, 0, 0);
#else
  *(float4*)l = *(const float4*)g;
#endif
}

__device__ __forceinline__ void async_stage_wait() {
#ifdef USE_ASYNC_LDS
#if __has_builtin(__builtin_amdgcn_s_wait_asynccnt)
  __builtin_amdgcn_s_wait_asynccnt(0);
#else
  asm volatile("s_wait_asynccnt 0x0" ::: "memory");
#endif
#endif
}

// f32 WMMA: D(16x16) = A(16x4) * B(4x16) + C ; exact f32 math (RNE)
__device__ __forceinline__ v8f wmma_f32_4(v2f a, v2f b, v8f c) {
  return __builtin_amdgcn_wmma_f32_16x16x4_f32(false, a, false, b, (short)0, c,
                                               false, false);
}

// B fragment: lanes 0-15 -> rows k,k+1 ; lanes 16-31 -> rows k+2,k+3 (k pre-offset)
__device__ __forceinline__ v2f load_bfrag(const float* __restrict__ W, int k, int c) {
  v2f b;
  b.x = W[(size_t)k * HDIM + c];
  b.y = W[(size_t)(k + 1) * HDIM + c];
  return b;
}

// ---------------------------------------------------------------------------
// Scatter-mean accumulation: sum[dst] += src[srcIdx], cnt[dst] += 1
// One thread per (edge, 4-float chunk); 32 chunks cover H=128.
// ---------------------------------------------------------------------------
__global__ __launch_bounds__(256) void scatter_accum_kernel(
    const float* __restrict__ src, const int* __restrict__ ei, int E,
    float* __restrict__ sum, float* __restrict__ cnt) {
  int gid = blockIdx.x * 256 + threadIdx.x;
  int e = gid >> 5;
  if (e >= E) return;
  int part = gid & 31;
  int s = ei[e];
  int d = ei[E + e];
  const float4 v = *(const float4*)&src[(size_t)s * HDIM + part * 4];
  float* dp = &sum[(size_t)d * HDIM + part * 4];
  atomicAdd(dp + 0, v.x);
  atomicAdd(dp + 1, v.y);
  atomicAdd(dp + 2, v.z);
  atomicAdd(dp + 3, v.w);
  if (part == 0) atomicAdd(&cnt[d], 1.0f);
}

// ---------------------------------------------------------------------------
// Input projection: out = X[N,K] @ W[K,128] + b   (K = 128 or 64)
// 64 rows per block; each wave: 1 N-tile x 4 M-tiles (B fragments reused 4x).
// ---------------------------------------------------------------------------
template <int K>
__global__ __launch_bounds__(256) void proj_kernel(
    const float* __restrict__ X, const float* __restrict__ W,
    const float* __restrict__ bias, float* __restrict__ out, int N) {
  __shared__ __align__(16) float sX[64][K + 4];
  const int r0 = blockIdx.x * 64;
  const int t = threadIdx.x;
  for (int i = t; i < 64 * (K / 4); i += 256) {
    int r = i / (K / 4), c4 = (i % (K / 4)) * 4;
    int rr = (r0 + r < N) ? (r0 + r) : (N - 1);
    g2lds_f4(&X[(size_t)rr * K + c4], &sX[r][c4]);
  }
  async_stage_wait();
  __syncthreads();

  const int wave = t >> 5, lane = t & 31;
  const int n16 = lane & 15, half = lane >> 4, ka = half * 2;
  const int c0 = wave * 16;
  v8f acc[4];
#pragma unroll
  for (int mt = 0; mt < 4; mt++) acc[mt] = (v8f){0.f, 0.f, 0.f, 0.f, 0.f, 0.f, 0.f, 0.f};
  for (int k = 0; k < K; k += 4) {
    v2f b = load_bfrag(W, k + ka, c0 + n16);
#pragma unroll
    for (int mt = 0; mt < 4; mt++) {
      int m = mt * 16 + n16;
      v2f a; a.x = sX[m][k + ka]; a.y = sX[m][k + ka + 1];
      acc[mt] = wmma_f32_4(a, b, acc[mt]);
    }
  }
  const float bb = bias[c0 + n16];
#pragma unroll
  for (int mt = 0; mt < 4; mt++) {
#pragma unroll
    for (int i = 0; i < 8; i++) {
      int rr = r0 + mt * 16 + i + half * 8;
      if (rr < N) out[(size_t)rr * HDIM + c0 + n16] = acc[mt][i] + bb;
    }
  }
}

// ---------------------------------------------------------------------------
// Shared epilogue: LayerNorm + exact-erf GELU (or passthrough) over a
// 64x128 LDS tile. 16 threads per row, 4 row-passes.
// ---------------------------------------------------------------------------
__device__ __forceinline__ void epilogue_ln_gelu(
    const float (*sOut)[HDIM + 4], const float* __restrict__ ln_g,
    const float* __restrict__ ln_b, int do_ln, float* __restrict__ out, int r0,
    int N, int t) {
  const int q = t & 15;
#pragma unroll
  for (int rp = 0; rp < 4; rp++) {
    int r = rp * 16 + (t >> 4);
    float vals[8];
    float s = 0.f, s2 = 0.f;
#pragma unroll
    for (int j = 0; j < 8; j++) {
      float x = sOut[r][q * 8 + j];
      vals[j] = x; s += x; s2 += x * x;
    }
    for (int off = 1; off < 16; off <<= 1) {
      s += __shfl_xor(s, off, 32);
      s2 += __shfl_xor(s2, off, 32);
    }
    if (r0 + r < N) {
      float* orow = &out[(size_t)(r0 + r) * HDIM];
      if (do_ln) {
        float mean = s * (1.0f / HDIM);
        float var = s2 * (1.0f / HDIM) - mean * mean;
        float rstd = rsqrtf(var + 1e-5f);
#pragma unroll
        for (int j = 0; j < 8; j++) {
          int c = q * 8 + j;
          float x = (vals[j] - mean) * rstd * ln_g[c] + ln_b[c];
          orow[c] = 0.5f * x * (1.0f + erff(x * 0.70710678118654752f));
        }
      } else {
#pragma unroll
        for (int j = 0; j < 8; j++) orow[q * 8 + j] = vals[j];
      }
    }
  }
}

// ---------------------------------------------------------------------------
// Paper update: new_hp = 0.5*((meanW@WlW + blW + hp@WrW) + (meanC@WlC + blC + hp@WrC))
// 64 rows/block; mean division folded into A-fragment load.
// ---------------------------------------------------------------------------
__global__ __launch_bounds__(256) void paper_update_kernel(
    const float* __restrict__ hp, const float* __restrict__ sumW,
    const float* __restrict__ cntW, const float* __restrict__ sumC,
    const float* __restrict__ cntC, const float* __restrict__ WlW,
    const float* __restrict__ WrW, const float* __restrict__ blW,
    const float* __restrict__ WlC, const float* __restrict__ WrC,
    const float* __restrict__ blC, const float* __restrict__ ln_g,
    const float* __restrict__ ln_b, int do_ln, float* __restrict__ out,
    int NP) {
  __shared__ __align__(16) float sHp[64][HDIM + 4];
  __shared__ __align__(16) float sMw[64][HDIM + 4];
  __shared__ __align__(16) float sMc[64][HDIM + 4];
  __shared__ __align__(16) float sOut[64][HDIM + 4];
  __shared__ float sInvW[64], sInvC[64];

  const int r0 = blockIdx.x * 64;
  const int t = threadIdx.x;
  if (t < 64) {
    int rr = (r0 + t < NP) ? (r0 + t) : (NP - 1);
    sInvW[t] = 1.0f / fmaxf(cntW[rr], 1.0f);
    sInvC[t] = 1.0f / fmaxf(cntC[rr], 1.0f);
  }
  for (int i = t; i < 64 * 32; i += 256) {
    int r = i >> 5, c4 = (i & 31) * 4;
    int rr = (r0 + r < NP) ? (r0 + r) : (NP - 1);
    size_t base = (size_t)rr * HDIM + c4;
    g2lds_f4(&hp[base], &sHp[r][c4]);
    g2lds_f4(&sumW[base], &sMw[r][c4]);
    g2lds_f4(&sumC[base], &sMc[r][c4]);
  }
  async_stage_wait();
  __syncthreads();

  const int wave = t >> 5, lane = t & 31;
  const int n16 = lane & 15, half = lane >> 4, ka = half * 2;
  const int c0 = wave * 16;
  v8f acc[4];
  float iw[4], icc[4];
#pragma unroll
  for (int mt = 0; mt < 4; mt++) {
    acc[mt] = (v8f){0.f, 0.f, 0.f, 0.f, 0.f, 0.f, 0.f, 0.f};
    iw[mt] = sInvW[mt * 16 + n16];
    icc[mt] = sInvC[mt * 16 + n16];
  }
  for (int k = 0; k < HDIM; k += 4) {
    v2f bWl = load_bfrag(WlW, k + ka, c0 + n16);
    v2f bWr = load_bfrag(WrW, k + ka, c0 + n16);
    v2f bCl = load_bfrag(WlC, k + ka, c0 + n16);
    v2f bCr = load_bfrag(WrC, k + ka, c0 + n16);
#pragma unroll
    for (int mt = 0; mt < 4; mt++) {
      int m = mt * 16 + n16;
      v2f aH; aH.x = sHp[m][k + ka];           aH.y = sHp[m][k + ka + 1];
      v2f aW; aW.x = sMw[m][k + ka] * iw[mt];  aW.y = sMw[m][k + ka + 1] * iw[mt];
      v2f aC; aC.x = sMc[m][k + ka] * icc[mt]; aC.y = sMc[m][k + ka + 1] * icc[mt];
      acc[mt] = wmma_f32_4(aW, bWl, acc[mt]);
      acc[mt] = wmma_f32_4(aH, bWr, acc[mt]);
      acc[mt] = wmma_f32_4(aC, bCl, acc[mt]);
      acc[mt] = wmma_f32_4(aH, bCr, acc[mt]);
    }
  }
  const float bsum = blW[c0 + n16] + blC[c0 + n16];
#pragma unroll
  for (int mt = 0; mt < 4; mt++) {
#pragma unroll
    for (int i = 0; i < 8; i++) {
      sOut[mt * 16 + i + half * 8][c0 + n16] = 0.5f * (acc[mt][i] + bsum);
    }
  }
  __syncthreads();
  epilogue_ln_gelu(sOut, ln_g, ln_b, do_ln, out, r0, NP, t);
}

// ---------------------------------------------------------------------------
// Author update: new_ha = meanWB@Wl + bl + ha@Wr ; optional LN+GELU.
// ---------------------------------------------------------------------------
__global__ __launch_bounds__(256) void author_update_kernel(
    const float* __restrict__ ha, const float* __restrict__ sumWB,
    const float* __restrict__ cntWB, const float* __restrict__ Wl,
    const float* __restrict__ Wr, const float* __restrict__ bl,
    const float* __restrict__ ln_g, const float* __restrict__ ln_b, int do_ln,
    float* __restrict__ out, int NA) {
  __shared__ __align__(16) float sHa[64][HDIM + 4];
  __shared__ __align__(16) float sMn[64][HDIM + 4];
  __shared__ __align__(16) float sOut[64][HDIM + 4];
  __shared__ float sInv[64];

  const int r0 = blockIdx.x * 64;
  const int t = threadIdx.x;
  if (t < 64) {
    int rr = (r0 + t < NA) ? (r0 + t) : (NA - 1);
    sInv[t] = 1.0f / fmaxf(cntWB[rr], 1.0f);
  }
  for (int i = t; i < 64 * 32; i += 256) {
    int r = i >> 5, c4 = (i & 31) * 4;
    int rr = (r0 + r < NA) ? (r0 + r) : (NA - 1);
    size_t base = (size_t)rr * HDIM + c4;
    g2lds_f4(&ha[base], &sHa[r][c4]);
    g2lds_f4(&sumWB[base], &sMn[r][c4]);
  }
  async_stage_wait();
  __syncthreads();

  const int wave = t >> 5, lane = t & 31;
  const int n16 = lane & 15, half = lane >> 4, ka = half * 2;
  const int c0 = wave * 16;
  v8f acc[4];
  float iv[4];
#pragma unroll
  for (int mt = 0; mt < 4; mt++) {
    acc[mt] = (v8f){0.f, 0.f, 0.f, 0.f, 0.f, 0.f, 0.f, 0.f};
    iv[mt] = sInv[mt * 16 + n16];
  }
  for (int k = 0; k < HDIM; k += 4) {
    v2f bL = load_bfrag(Wl, k + ka, c0 + n16);
    v2f bR = load_bfrag(Wr, k + ka, c0 + n16);
#pragma unroll
    for (int mt = 0; mt < 4; mt++) {
      int m = mt * 16 + n16;
      v2f aH; aH.x = sHa[m][k + ka];          aH.y = sHa[m][k + ka + 1];
      v2f aM; aM.x = sMn[m][k + ka] * iv[mt]; aM.y = sMn[m][k + ka + 1] * iv[mt];
      acc[mt] = wmma_f32_4(aM, bL, acc[mt]);
      acc[mt] = wmma_f32_4(aH, bR, acc[mt]);
    }
  }
  const float bb = bl[c0 + n16];
#pragma unroll
  for (int mt = 0; mt < 4; mt++) {
#pragma unroll
    for (int i = 0; i < 8; i++) {
      sOut[mt * 16 + i + half * 8][c0 + n16] = acc[mt][i] + bb;
    }
  }
  __syncthreads();
  epilogue_ln_gelu(sOut, ln_g, ln_b, do_ln, out, r0, NA, t);
}

// ---------------------------------------------------------------------------
// Host side
// ---------------------------------------------------------------------------
struct SageP { const float *Wl, *Wr, *bl; };
struct LayerP {
  SageP writes, written_by, cites;
  const float *lnp_g = nullptr, *lnp_b = nullptr;
  const float *lna_g = nullptr, *lna_b = nullptr;
};

extern "C" void kernel_launch(void* const* d_in, const int* in_sizes, int n_in,
                              void* d_out, int out_size, void* d_ws,
                              size_t ws_size, hipStream_t stream) {
  (void)n_in; (void)out_size; (void)ws_size;
  const float* x_paper = (const float*)d_in[0];
  const float* x_author = (const float*)d_in[1];
  const int* ei_writes = (const int*)d_in[2];
  const int* ei_written = (const int*)d_in[3];
  const int* ei_cites = (const int*)d_in[4];
  const int NP = in_sizes[0] / 128;  // 100000
  const int NA = in_sizes[1] / 64;   // 50000
  const int Ew = in_sizes[2] / 2;
  const int Ewb = in_sizes[3] / 2;
  const int Ec = in_sizes[4] / 2;

  auto F = [&](int i) { return (const float*)d_in[i]; };
  const float *ppW, *ppb, *paW, *pab;
  LayerP L[3];
  const bool sorted_order = (in_sizes[6] == 16384);  // jax-sorted vs insertion
  if (sorted_order) {
    int k = 5;
    for (int i = 0; i < 3; i++) {
      L[i].cites = {F(k), F(k + 1), F(k + 2)}; k += 3;          // Wl, Wr, bl
      if (i < 2) {
        L[i].lna_b = F(k); L[i].lna_g = F(k + 1);
        L[i].lnp_b = F(k + 2); L[i].lnp_g = F(k + 3); k += 4;
      }
      L[i].writes = {F(k), F(k + 1), F(k + 2)}; k += 3;
      L[i].written_by = {F(k), F(k + 1), F(k + 2)}; k += 3;
    }
    paW = F(k); pab = F(k + 1); ppW = F(k + 2); ppb = F(k + 3);
  } else {
    ppW = F(5); ppb = F(6); paW = F(7); pab = F(8);
    int k = 9;
    for (int i = 0; i < 3; i++) {
      L[i].writes = {F(k), F(k + 2), F(k + 1)}; k += 3;         // Wl, bl, Wr
      L[i].written_by = {F(k), F(k + 2), F(k + 1)}; k += 3;
      L[i].cites = {F(k), F(k + 2), F(k + 1)}; k += 3;
      if (i < 2) {
        L[i].lnp_g = F(k); L[i].lnp_b = F(k + 1);
        L[i].lna_g = F(k + 2); L[i].lna_b = F(k + 3); k += 4;
      }
    }
  }

  // Workspace carve-up
  float* ws = (float*)d_ws;
  const size_t NPH = (size_t)NP * HDIM, NAH = (size_t)NA * HDIM;
  float* hpbuf[2]; float* habuf[2];
  hpbuf[0] = ws; ws += NPH;
  hpbuf[1] = ws; ws += NPH;
  habuf[0] = ws; ws += NAH;
  habuf[1] = ws; ws += NAH;
  float* sumW = ws; ws += NPH;
  float* sumC = ws; ws += NPH;
  float* sumWB = ws; ws += NAH;
  float* cntW = ws; ws += NP;
  float* cntC = ws; ws += NP;
  float* cntWB = ws; ws += NA;

  float* out_hp = (float*)d_out;
  float* out_ha = out_hp + NPH;

  const int gp = (NP + 63) / 64, ga = (NA + 63) / 64;

  // Projections
  proj_kernel<128><<<gp, 256, 0, stream>>>(x_paper, ppW, ppb, hpbuf[0], NP);
  proj_kernel<64><<<ga, 256, 0, stream>>>(x_author, paW, pab, habuf[0], NA);

  int hpi = 0, hai = 0;
  for (int i = 0; i < 3; i++) {
    (void)hipMemsetAsync(sumW, 0, NPH * sizeof(float), stream);
    (void)hipMemsetAsync(sumC, 0, NPH * sizeof(float), stream);
    (void)hipMemsetAsync(sumWB, 0, NAH * sizeof(float), stream);
    (void)hipMemsetAsync(cntW, 0, NP * sizeof(float), stream);
    (void)hipMemsetAsync(cntC, 0, NP * sizeof(float), stream);
    (void)hipMemsetAsync(cntWB, 0, NA * sizeof(float), stream);

    const float* hp_cur = hpbuf[hpi];
    const float* ha_cur = habuf[hai];
    scatter_accum_kernel<<<(Ew * 32 + 255) / 256, 256, 0, stream>>>(
        ha_cur, ei_writes, Ew, sumW, cntW);
    scatter_accum_kernel<<<(Ec * 32 + 255) / 256, 256, 0, stream>>>(
        hp_cur, ei_cites, Ec, sumC, cntC);
    scatter_accum_kernel<<<(Ewb * 32 + 255) / 256, 256, 0, stream>>>(
        hp_cur, ei_written, Ewb, sumWB, cntWB);

    float* hp_out = (i == 2) ? out_hp : hpbuf[1 - hpi];
    float* ha_out = (i == 2) ? out_ha : habuf[1 - hai];
    const int do_ln = (i < 2) ? 1 : 0;

    paper_update_kernel<<<gp, 256, 0, stream>>>(
        hp_cur, sumW, cntW, sumC, cntC,
        L[i].writes.Wl, L[i].writes.Wr, L[i].writes.bl,
        L[i].cites.Wl, L[i].cites.Wr, L[i].cites.bl,
        L[i].lnp_g, L[i].lnp_b, do_ln, hp_out, NP);
    author_update_kernel<<<ga, 256, 0, stream>>>(
        ha_cur, sumWB, cntWB,
        L[i].written_by.Wl, L[i].written_by.Wr, L[i].written_by.bl,
        L[i].lna_g, L[i].lna_b, do_ln, ha_out, NA);

    hpi = 1 - hpi;
    hai = 1 - hai;
  }
}